// Energy_based_Debias_Loss_18081812316261
// MI455X (gfx1250) — compile-verified
//
#include <hip/hip_runtime.h>
#include <hip/hip_bf16.h>
#include <cstdint>

#define NEGV  (-1e30f)
#define BS    512
#define NWAVE (BS / 32)

__device__ __forceinline__ float wave_max(float v) {
#pragma unroll
    for (int off = 16; off; off >>= 1) v = fmaxf(v, __shfl_xor(v, off, 32));
    return v;
}
__device__ __forceinline__ float wave_sum(float v) {
#pragma unroll
    for (int off = 16; off; off >>= 1) v += __shfl_xor(v, off, 32);
    return v;
}

// Reduce two maxima with a single barrier sequence.
__device__ __forceinline__ void block_max2(float& a, float& b, float* sa, float* sb, int tid) {
    a = wave_max(a);
    b = wave_max(b);
    if ((tid & 31) == 0) { sa[tid >> 5] = a; sb[tid >> 5] = b; }
    __syncthreads();
    if (tid < 32) {
        float wa = (tid < NWAVE) ? sa[tid] : -INFINITY;
        float wb = (tid < NWAVE) ? sb[tid] : -INFINITY;
        wa = wave_max(wa);
        wb = wave_max(wb);
        if (tid == 0) { sa[0] = wa; sb[0] = wb; }
    }
    __syncthreads();
    a = sa[0]; b = sb[0];
    __syncthreads();
}

// Reduce two sums with a single barrier sequence.
__device__ __forceinline__ void block_sum2(float& a, float& b, float* sa, float* sb, int tid) {
    a = wave_sum(a);
    b = wave_sum(b);
    if ((tid & 31) == 0) { sa[tid >> 5] = a; sb[tid >> 5] = b; }
    __syncthreads();
    if (tid < 32) {
        float wa = (tid < NWAVE) ? sa[tid] : 0.0f;
        float wb = (tid < NWAVE) ? sb[tid] : 0.0f;
        wa = wave_sum(wa);
        wb = wave_sum(wb);
        if (tid == 0) { sa[0] = wa; sb[0] = wb; }
    }
    __syncthreads();
    a = sa[0]; b = sb[0];
    __syncthreads();
}

// Combine per-thread (running-max, running-scaled-sum) logsumexp states across the block.
__device__ __forceinline__ void block_lse(float& m, float& s, float* sm, float* ss, int tid) {
#pragma unroll
    for (int off = 16; off; off >>= 1) {
        float om = __shfl_xor(m, off, 32);
        float os = __shfl_xor(s, off, 32);
        float nm = fmaxf(m, om);
        s = s * __expf(m - nm) + os * __expf(om - nm);
        m = nm;
    }
    if ((tid & 31) == 0) { sm[tid >> 5] = m; ss[tid >> 5] = s; }
    __syncthreads();
    if (tid < 32) {
        float mm = (tid < NWAVE) ? sm[tid] : -INFINITY;
        float sc = (tid < NWAVE) ? ss[tid] : 0.0f;
#pragma unroll
        for (int off = 16; off; off >>= 1) {
            float om = __shfl_xor(mm, off, 32);
            float os = __shfl_xor(sc, off, 32);
            float nm = fmaxf(mm, om);
            sc = sc * __expf(mm - nm) + os * __expf(om - nm);
            mm = nm;
        }
        if (tid == 0) { sm[0] = mm; ss[0] = sc; }
    }
    __syncthreads();
    m = sm[0]; s = ss[0];
    __syncthreads();
}

__global__ __launch_bounds__(BS) void ebd_row_kernel(
    const float* __restrict__ logits,
    const int*   __restrict__ targets,
    const float* __restrict__ U,
    const float* __restrict__ class_bias,
    float* __restrict__ row_nll,
    int C)
{
    __shared__ __align__(16) float slog[32000];   // one full logits row (128 KB)
    __shared__ float sm[NWAVE], ss[NWAVE];
    __shared__ float s_zt;

    const int tid = threadIdx.x;
    const int row = blockIdx.x;
    const int C4  = C >> 2;
    const float* rowp = logits + (size_t)row * (size_t)C;
    const float* urow = U      + (size_t)row * (size_t)C;

    // ---- Phase A: async DMA the logits row into LDS (ASYNCcnt path) ----
    uint32_t lds_base = (uint32_t)(uintptr_t)(&slog[0]);
    for (int j4 = tid; j4 < C4; j4 += BS) {
        uint32_t loff = lds_base + (uint32_t)j4 * 16u;
        unsigned long long gp = (unsigned long long)(rowp + (size_t)j4 * 4);
        asm volatile("global_load_async_to_lds_b128 %0, %1, off"
                     :: "v"(loff), "v"(gp) : "memory");
    }
    // Warm L2 for the U row (consumed only in pass 2) while the DMA + pass 1 run.
    // global_prefetch_b8: no counters, fetches whole cachelines; 128 B stride.
    {
        const char* up = (const char*)urow;
        const int rowbytes = C * 4;
        for (int b = tid * 128; b < rowbytes; b += BS * 128)
            __builtin_prefetch(up + b, 0, 1);
    }
    asm volatile("s_wait_asynccnt 0x0" ::: "memory");
    __syncthreads();

    const int t = targets[row];
    const float4* slog4 = (const float4*)slog;

    // ---- Pass 1a: row max (full + target-masked), served from LDS ----
    float lmax = -INFINITY, lmax_ex = -INFINITY;
    for (int j4 = tid; j4 < C4; j4 += BS) {
        float4 v = slog4[j4];
        int j = j4 << 2;
        lmax = fmaxf(lmax, fmaxf(fmaxf(v.x, v.y), fmaxf(v.z, v.w)));
        float a = (j + 0 == t) ? NEGV : v.x;
        float b = (j + 1 == t) ? NEGV : v.y;
        float c = (j + 2 == t) ? NEGV : v.z;
        float d = (j + 3 == t) ? NEGV : v.w;
        lmax_ex = fmaxf(lmax_ex, fmaxf(fmaxf(a, b), fmaxf(c, d)));
    }
    float m = lmax, me = lmax_ex;
    block_max2(m, me, sm, ss, tid);

    // ---- Pass 1b: sum of exps (full + masked) ----
    float s = 0.0f, se = 0.0f;
    for (int j4 = tid; j4 < C4; j4 += BS) {
        float4 v = slog4[j4];
        int j = j4 << 2;
        s  += __expf(v.x - m) + __expf(v.y - m) + __expf(v.z - m) + __expf(v.w - m);
        se += ((j + 0 == t) ? 0.0f : __expf(v.x - me))
            + ((j + 1 == t) ? 0.0f : __expf(v.y - me))
            + ((j + 2 == t) ? 0.0f : __expf(v.z - me))
            + ((j + 3 == t) ? 0.0f : __expf(v.w - me));
    }
    float S = s, Se = se;
    block_sum2(S, Se, sm, ss, tid);

    float lse  = m  + __logf(S);    // logsumexp(logits)
    float lsee = me + __logf(Se);   // logsumexp(masked)
    float energy = -lse;
    float be     = -lsee;
    float gb = ((be > 0.0f) ? 0.0f : be) / energy;   // gumble_beta (BETA = 1)

    // ---- Pass 2: online logsumexp of z = l + gb*g(U) + log(class_bias) ----
    const float4* U4  = (const float4*)urow;
    const float4* CB4 = (const float4*)class_bias;
    float mz = -INFINITY, sz = 0.0f;
    for (int j4 = tid; j4 < C4; j4 += BS) {
        float4 l  = slog4[j4];
        float4 u  = U4[j4];
        float4 cb = CB4[j4];
        float z0 = l.x + gb * (-__logf(-__logf(u.x + 1e-10f) + 1e-10f)) + __logf(cb.x + 1e-12f);
        float z1 = l.y + gb * (-__logf(-__logf(u.y + 1e-10f) + 1e-10f)) + __logf(cb.y + 1e-12f);
        float z2 = l.z + gb * (-__logf(-__logf(u.z + 1e-10f) + 1e-10f)) + __logf(cb.z + 1e-12f);
        float z3 = l.w + gb * (-__logf(-__logf(u.w + 1e-10f) + 1e-10f)) + __logf(cb.w + 1e-12f);
        if ((t >> 2) == j4) {
            int k = t & 3;
            s_zt = (k == 0) ? z0 : (k == 1) ? z1 : (k == 2) ? z2 : z3;
        }
        float h = fmaxf(fmaxf(z0, z1), fmaxf(z2, z3));
        float p = __expf(z0 - h) + __expf(z1 - h) + __expf(z2 - h) + __expf(z3 - h);
        float nm = fmaxf(mz, h);
        sz = sz * __expf(mz - nm) + p * __expf(h - nm);
        mz = nm;
    }
    block_lse(mz, sz, sm, ss, tid);    // contains barriers -> s_zt visible
    if (tid == 0) row_nll[row] = (mz + __logf(sz)) - s_zt;
}

__global__ __launch_bounds__(1024) void mean_kernel(
    const float* __restrict__ x, float* __restrict__ out, int n)
{
    __shared__ float red[32];
    int tid = threadIdx.x;
    float s = 0.0f;
    for (int i = tid; i < n; i += 1024) s += x[i];
    s = wave_sum(s);
    if ((tid & 31) == 0) red[tid >> 5] = s;
    __syncthreads();
    if (tid < 32) {
        float v = red[tid];          // exactly 32 waves -> fully populated
        v = wave_sum(v);
        if (tid == 0) out[0] = v / (float)n;
    }
}

extern "C" void kernel_launch(void* const* d_in, const int* in_sizes, int n_in,
                              void* d_out, int out_size, void* d_ws, size_t ws_size,
                              hipStream_t stream) {
    const float* logits     = (const float*)d_in[0];
    const int*   targets    = (const int*)  d_in[1];
    const float* U          = (const float*)d_in[2];
    const float* class_bias = (const float*)d_in[3];
    // d_in[4] (adaptive_marg_coef) and d_in[5] (w_norm) are unused by the reference.

    const int N = in_sizes[1];   // 4096 rows
    const int C = in_sizes[3];   // 32000 classes (LDS row buffer sized for this)

    float* row_nll = (float*)d_ws;   // N floats of scratch

    ebd_row_kernel<<<N, BS, 0, stream>>>(logits, targets, U, class_bias, row_nll, C);
    mean_kernel<<<1, 1024, 0, stream>>>(row_nll, (float*)d_out, N);
}